// MatchingMetric_75857712382593
// MI455X (gfx1250) — compile-verified
//
#include <hip/hip_runtime.h>

// Problem constants from the reference.
#define BATCH   64
#define NT      256
#define NP      900
#define NP4     (NP / 4)     // 225 float4 groups per row
#define TROWS   16           // target rows per block
#define THREADS 256          // 8 wave32 waves

typedef float f32x4 __attribute__((ext_vector_type(4)));
typedef int   i32x4 __attribute__((vector_size(16)));

// Pointer types matching the async-to-LDS builtin's expected signature:
// (int4 addrspace(1)*, int4 addrspace(3)*, imm offset, imm cpol)
typedef __attribute__((address_space(1))) i32x4* as1_i32x4_p;
typedef __attribute__((address_space(3))) i32x4* as3_i32x4_p;

__global__ __launch_bounds__(THREADS)
void masked_iou_kernel(const float* __restrict__ bbox,    // [B, NT, 4] (y1,x1,y2,x2)
                       const float* __restrict__ preds,   // [B, NP, 4]
                       const float* __restrict__ mask,    // [B, NT, NP]
                       float* __restrict__ out)           // [B, NT, NP]
{
    // AoS staging target for async copies, then SoA for conflict-free reads.
    __shared__ __align__(16) f32x4 sAoS[NP];      // 14400 B
    __shared__ __align__(16) float sPy1[NP];      //  3600 B
    __shared__ __align__(16) float sPx1[NP];
    __shared__ __align__(16) float sPy2[NP];
    __shared__ __align__(16) float sPx2[NP];
    __shared__ __align__(16) f32x4 sT[TROWS];     //   256 B

    const int tid = threadIdx.x;
    const int blk = blockIdx.x;
    const int b   = blk / (NT / TROWS);
    const int t0  = (blk % (NT / TROWS)) * TROWS;

    const float* gP = preds + (size_t)b * NP * 4;
    const float* gT = bbox  + ((size_t)b * NT + t0) * 4;

#if defined(__gfx1250__) && __has_builtin(__builtin_amdgcn_global_load_async_to_lds_b128)
    // CDNA5 async global->LDS DMA path (ASYNCcnt).
    for (int j = tid; j < NP; j += THREADS) {
        __builtin_amdgcn_global_load_async_to_lds_b128(
            (as1_i32x4_p)(gP + (size_t)j * 4),
            (as3_i32x4_p)&sAoS[j],
            /*offset=*/0, /*cpol=*/0);
    }
    if (tid < TROWS) {
        __builtin_amdgcn_global_load_async_to_lds_b128(
            (as1_i32x4_p)(gT + (size_t)tid * 4),
            (as3_i32x4_p)&sT[tid],
            /*offset=*/0, /*cpol=*/0);
    }
  #if __has_builtin(__builtin_amdgcn_s_wait_asynccnt)
    __builtin_amdgcn_s_wait_asynccnt(0);
  #else
    asm volatile("s_wait_asynccnt 0" ::: "memory");
  #endif
#else
    // Fallback: plain cooperative vector copy.
    for (int j = tid; j < NP; j += THREADS)
        sAoS[j] = *(const f32x4*)(gP + (size_t)j * 4);
    if (tid < TROWS)
        sT[tid] = *(const f32x4*)(gT + (size_t)tid * 4);
#endif

    // AoS -> SoA transpose (each thread handles the same j's it loaded, so its
    // own asynccnt wait covers the data; barrier publishes to other waves).
    for (int j = tid; j < NP; j += THREADS) {
        f32x4 v = sAoS[j];
        sPy1[j] = v.x; sPx1[j] = v.y; sPy2[j] = v.z; sPx2[j] = v.w;
    }
    __syncthreads();

    const size_t base = ((size_t)b * NT + t0) * NP;

    // TROWS * NP4 = 3600 float4 output groups per block.
    for (int idx = tid; idx < TROWS * NP4; idx += THREADS) {
        const int row = idx / NP4;
        const int pq  = idx - row * NP4;

        const f32x4 tb = sT[row];                       // broadcast read
        const float t_h = fmaxf(tb.z - tb.x, 0.0f);
        const float t_w = fmaxf(tb.w - tb.y, 0.0f);
        const float area_t = t_h * t_w;

        // Conflict-free ds_load_b128: 16B per lane, 16B lane stride.
        const f32x4 py1 = *(const f32x4*)&sPy1[pq * 4];
        const f32x4 px1 = *(const f32x4*)&sPx1[pq * 4];
        const f32x4 py2 = *(const f32x4*)&sPy2[pq * 4];
        const f32x4 px2 = *(const f32x4*)&sPx2[pq * 4];

        const size_t off = base + (size_t)row * NP + (size_t)pq * 4;
        const f32x4 m = __builtin_nontemporal_load((const f32x4*)(mask + off));

        f32x4 r;
        #pragma unroll
        for (int k = 0; k < 4; ++k) {
            const float q1 = ((const float*)&py1)[k];
            const float q2 = ((const float*)&px1)[k];
            const float q3 = ((const float*)&py2)[k];
            const float q4 = ((const float*)&px2)[k];
            const float area_p = fmaxf(q3 - q1, 0.0f) * fmaxf(q4 - q2, 0.0f);
            const float ih = fminf(tb.z, q3) - fmaxf(tb.x, q1);
            const float iw = fminf(tb.w, q4) - fmaxf(tb.y, q2);
            const float inter = fmaxf(ih, 0.0f) * fmaxf(iw, 0.0f);
            const float uni = area_t + area_p - inter;
#if __has_builtin(__builtin_amdgcn_rcpf)
            const float iou = (uni > 0.0f) ? inter * __builtin_amdgcn_rcpf(uni) : 0.0f;
#else
            const float iou = (uni > 0.0f) ? inter / uni : 0.0f;
#endif
            ((float*)&r)[k] = iou;
        }
        r.x *= m.x; r.y *= m.y; r.z *= m.z; r.w *= m.w;

        __builtin_nontemporal_store(r, (f32x4*)(out + off));
    }
}

extern "C" void kernel_launch(void* const* d_in, const int* in_sizes, int n_in,
                              void* d_out, int out_size, void* d_ws, size_t ws_size,
                              hipStream_t stream) {
    (void)in_sizes; (void)n_in; (void)out_size; (void)d_ws; (void)ws_size;
    const float* bbox  = (const float*)d_in[0];   // [64, 256, 4]
    const float* preds = (const float*)d_in[1];   // [64, 900, 4]
    const float* mask  = (const float*)d_in[2];   // [64, 256, 900]
    float* out = (float*)d_out;                   // [64, 256, 900]

    dim3 grid(BATCH * (NT / TROWS));              // 1024 blocks
    dim3 block(THREADS);                          // 8 wave32 waves
    masked_iou_kernel<<<grid, block, 0, stream>>>(bbox, preds, mask, out);
}